// HungarianMatcher_70153995813079
// MI455X (gfx1250) — compile-verified
//
#include <hip/hip_runtime.h>

// Hungarian matcher cost matrix for MI455X (gfx1250, wave32).
//   C[bq, g] = -logits[bq, label[g]] + L1(box_bq, box_g) - GIoU(box_bq, box_g)
//
// cost_class is computed EXACTLY with V_WMMA_F32_16X16X4_F32 as a one-hot
// matmul: A = logits tile (16xK fp32), B = onehot(labels) (Kx16 fp32).
// Every product is logit*1.0 or logit*0.0 -> no rounding vs the fp32 gather.
// The L1 + GIoU epilogue is VALU. The kernel is HBM-store bound (~92 MB out),
// so everything reused is staged once in LDS and stores stream coalesced.

typedef __attribute__((ext_vector_type(2))) float v2f;
typedef __attribute__((ext_vector_type(8))) float v8f;

#define BQ_TOT 14400   // B*Q = 16*900
#define G_TOT  1600    // total GT
#define NCLS   91
#define KPAD   92      // pad K to multiple of 4; label==91 never occurs
#define EPSF   1e-6f

__global__ __launch_bounds__(128)
void hungarian_cost_wmma(const float* __restrict__ logits,   // [BQ, 91]
                         const float* __restrict__ pboxes,   // [BQ, 4] cxcywh
                         const int*   __restrict__ glabels,  // [G]
                         const float* __restrict__ gboxes,   // [G, 4] cxcywh
                         float*       __restrict__ out)      // [BQ, G]
{
    __shared__ float sLog[16][KPAD];   // logits tile, col 91 zero-padded
    __shared__ float sPB[16][4];       // pred boxes (16 query rows)
    __shared__ float sGB[64][4];       // GT boxes   (64 cols)
    __shared__ int   sGL[64];          // GT labels  (64 cols)

    const int tid  = threadIdx.x;
    const int row0 = blockIdx.y * 16;   // query-row tile base
    const int col0 = blockIdx.x * 64;   // GT-col tile base

    // ---- stage tiles into LDS (coalesced global loads) ----
    for (int i = tid; i < 16 * NCLS; i += 128) {
        const int r = i / NCLS, c = i - r * NCLS;
        sLog[r][c] = logits[(row0 + r) * NCLS + c];
    }
    if (tid < 16) sLog[tid][NCLS] = 0.0f;       // K pad column
    if (tid < 64) {
        sPB[tid >> 2][tid & 3] = pboxes[(row0 + (tid >> 2)) * 4 + (tid & 3)];
        sGL[tid] = glabels[col0 + tid];
    }
    for (int i = tid; i < 64 * 4; i += 128) {
        sGB[i >> 2][i & 3] = gboxes[(col0 + (i >> 2)) * 4 + (i & 3)];
    }
    __syncthreads();

    // ---- per-wave 16x16 WMMA tile: one-hot matmul accumulates the gather ----
    const int wave  = tid >> 5;
    const int lane  = tid & 31;
    const int l16   = lane & 15;
    const int hi    = lane >> 4;             // 0: lanes 0-15, 1: lanes 16-31
    const int myCol = (wave << 4) + l16;     // 0..63 within block tile
    const int lab   = sGL[myCol];

    v8f acc = {};
    #pragma unroll
    for (int k0 = 0; k0 < KPAD; k0 += 4) {
        const int kb = k0 + 2 * hi;
        // A 16x4 f32: lane l<16 -> M=l, K={k0,k0+1}; lanes 16-31 -> K={k0+2,k0+3}
        v2f a = *(const v2f*)&sLog[l16][kb];          // ds_load_b64, 8B aligned
        // B 4x16 one-hot (lane = column N, symmetric K split)
        v2f b;
        b.x = (lab == kb    ) ? 1.0f : 0.0f;
        b.y = (lab == kb + 1) ? 1.0f : 0.0f;
        acc = __builtin_amdgcn_wmma_f32_16x16x4_f32(
            /*neg_a=*/false, a, /*neg_b=*/false, b,
            /*c_mod=*/(short)0, acc, /*reuse_a=*/false, /*reuse_b=*/false);
    }
    // acc[v] = logits[row0 + v + 8*hi, lab_of(col0 + myCol)]  (exact)

    // ---- epilogue: L1 + GIoU, fold in class cost, stream stores ----
    const float gcx = sGB[myCol][0], gcy = sGB[myCol][1];
    const float gw  = sGB[myCol][2], gh  = sGB[myCol][3];
    const float gx0 = gcx - 0.5f * gw, gy0 = gcy - 0.5f * gh;
    const float gx1 = gcx + 0.5f * gw, gy1 = gcy + 0.5f * gh;
    const float garea = (gx1 - gx0) * (gy1 - gy0);
    const int   ocol  = col0 + myCol;

    #pragma unroll
    for (int v = 0; v < 8; ++v) {
        const int m = v + (hi << 3);                 // C/D row within tile
        const float pcx = sPB[m][0], pcy = sPB[m][1];
        const float pw  = sPB[m][2], ph  = sPB[m][3];

        const float l1 = fabsf(pcx - gcx) + fabsf(pcy - gcy) +
                         fabsf(pw  - gw ) + fabsf(ph  - gh );

        const float px0 = pcx - 0.5f * pw, py0 = pcy - 0.5f * ph;
        const float px1 = pcx + 0.5f * pw, py1 = pcy + 0.5f * ph;
        const float parea = (px1 - px0) * (py1 - py0);

        float iw = fminf(px1, gx1) - fmaxf(px0, gx0); iw = fmaxf(iw, 0.0f);
        float ih = fminf(py1, gy1) - fmaxf(py0, gy0); ih = fmaxf(ih, 0.0f);
        const float inter = iw * ih;
        const float uni   = parea + garea - inter;
        const float iou   = inter / (uni + EPSF);

        float ew = fmaxf(px1, gx1) - fminf(px0, gx0); ew = fmaxf(ew, 0.0f);
        float eh = fmaxf(py1, gy1) - fminf(py0, gy0); eh = fmaxf(eh, 0.0f);
        const float earea = ew * eh;
        const float giou  = iou - (earea - uni) / (earea + EPSF);

        // C = cost_class + cost_bbox + cost_giou = -gather + L1 - giou
        out[(size_t)(row0 + m) * G_TOT + ocol] = l1 - acc[v] - giou;
    }
}

extern "C" void kernel_launch(void* const* d_in, const int* in_sizes, int n_in,
                              void* d_out, int out_size, void* d_ws, size_t ws_size,
                              hipStream_t stream) {
    (void)in_sizes; (void)n_in; (void)d_ws; (void)ws_size; (void)out_size;
    const float* logits  = (const float*)d_in[0];  // [16,900,91]
    const float* pboxes  = (const float*)d_in[1];  // [16,900,4]
    const int*   glabels = (const int*)d_in[2];    // [1600]
    const float* gboxes  = (const float*)d_in[3];  // [1600,4]
    float*       out     = (float*)d_out;          // [16,900,1600]

    dim3 grid(G_TOT / 64, BQ_TOT / 16);            // 25 x 900 blocks
    hungarian_cost_wmma<<<grid, 128, 0, stream>>>(logits, pboxes, glabels,
                                                  gboxes, out);
}